// BetweenClusterFC_62749472195389
// MI455X (gfx1250) — compile-verified
//
#include <hip/hip_runtime.h>
#include <hip/hip_bf16.h>

typedef float v2f __attribute__((ext_vector_type(2)));
typedef float v8f __attribute__((ext_vector_type(8)));

#define D_IN  256
#define D_OUT 128

// ---------------------------------------------------------------------------
// Transpose W (256x128) -> WT (128x256) so B fragments become contiguous
// float2 loads in the GEMM inner loop.
// ---------------------------------------------------------------------------
__global__ void wt_transpose_kernel(const float* __restrict__ W,
                                    float* __restrict__ WT) {
    int idx = blockIdx.x * blockDim.x + threadIdx.x;   // 0 .. 32767
    if (idx >= D_IN * D_OUT) return;
    int k = idx >> 7;        // row in W   (0..255)
    int n = idx & 127;       // col in W   (0..127)
    WT[n * D_IN + k] = W[idx];
}

// ---------------------------------------------------------------------------
// P = E (Mx256) @ W (256x128) + bias, via V_WMMA_F32_16X16X4_F32.
// One wave computes a 16(M) x 128(N) strip: A is read exactly once from HBM,
// B (WT, 128KB) is cache-resident. 64 k-steps x 8 n-tiles = 512 WMMAs/wave.
//
// Software-pipelined: the 9 fragment loads (1x A + 8x B) for k-step k+4 are
// issued into the alternate register buffer BEFORE the 8 WMMAs of k-step k,
// so the WMMAs cover the load latency instead of stalling on
// s_wait_loadcnt 0 before every matrix op.
//
// fp32 A 16x4 layout (ISA 7.12.2): lanes 0-15 hold M=0..15 {K=0,K=1},
//                                  lanes 16-31 hold M=0..15 {K=2,K=3}.
// fp32 B 4x16 layout (mirror):     lanes 0-15 hold N=0..15 {K=0,K=1},
//                                  lanes 16-31 hold N=0..15 {K=2,K=3}.
// fp32 C/D 16x16: VGPR r, lanes 0-15 -> M=r, N=lane; lanes 16-31 -> M=r+8.
// ---------------------------------------------------------------------------
__global__ void proj_gemm_wmma_kernel(const float* __restrict__ E,
                                      const float* __restrict__ WT,
                                      const float* __restrict__ bias,
                                      float* __restrict__ P,
                                      int strips) {
    int waveId = (int)((blockIdx.x * (unsigned)blockDim.x + threadIdx.x) >> 5);
    int lane   = (int)(threadIdx.x & 31u);
    if (waveId >= strips) return;            // wave-uniform: EXEC stays all-1s

    const int m0   = waveId << 4;            // first of 16 rows
    const int half = lane >> 4;              // 0: K=0,1   1: K=2,3
    const int l15  = lane & 15;

    const float* Arow = E  + (size_t)(m0 + l15) * D_IN + half * 2;
    const float* WTl  = WT + (size_t)l15 * D_IN + half * 2;

    v8f acc[8];
#pragma unroll
    for (int i = 0; i < 8; ++i) acc[i] = (v8f){0.f,0.f,0.f,0.f,0.f,0.f,0.f,0.f};

    // double-buffered fragments
    v2f a[2];
    v2f b[2][8];

    auto load_group = [&](int k0, int buf) {
        float2 av = *(const float2*)(Arow + k0);
        a[buf].x = av.x; a[buf].y = av.y;
#pragma unroll
        for (int nt = 0; nt < 8; ++nt) {
            float2 bv = *(const float2*)(WTl + nt * 16 * D_IN + k0);
            b[buf][nt].x = bv.x; b[buf][nt].y = bv.y;
        }
    };

    load_group(0, 0);

#pragma unroll 2
    for (int k0 = 0; k0 < D_IN - 4; k0 += 4) {
        const int cur = (k0 >> 2) & 1;
        load_group(k0 + 4, cur ^ 1);     // prefetch next k-step into other buf
#pragma unroll
        for (int nt = 0; nt < 8; ++nt) {
            acc[nt] = __builtin_amdgcn_wmma_f32_16x16x4_f32(
                false, a[cur], false, b[cur][nt],
                (short)0, acc[nt], false, false);
        }
    }
    {   // epilogue: last k-step (buffer index ((D_IN-4)>>2)&1 == 1)
        const int cur = ((D_IN - 4) >> 2) & 1;
#pragma unroll
        for (int nt = 0; nt < 8; ++nt) {
            acc[nt] = __builtin_amdgcn_wmma_f32_16x16x4_f32(
                false, a[cur], false, b[cur][nt],
                (short)0, acc[nt], false, false);
        }
    }

    const int mbase = m0 + half * 8;
#pragma unroll
    for (int nt = 0; nt < 8; ++nt) {
        float bb = bias[nt * 16 + l15];
#pragma unroll
        for (int r = 0; r < 8; ++r) {
            P[(size_t)(mbase + r) * D_OUT + nt * 16 + l15] = acc[nt][r] + bb;
        }
    }
}

// ---------------------------------------------------------------------------
// out[e] = dot(P1[from[e]], P2[to[e]]), 128-dim. One wave32 per edge:
// each lane loads one float4 of each row (gathers hit the 192MB L2 since
// P1+P2 ~= 102MB), then a 5-step shfl_xor butterfly reduction.
// ---------------------------------------------------------------------------
__global__ void edge_dot_kernel(const float* __restrict__ P1,
                                const float* __restrict__ P2,
                                const long long* __restrict__ edges,
                                float* __restrict__ out,
                                int nEdges) {
    int wave = (int)((blockIdx.x * (unsigned)blockDim.x + threadIdx.x) >> 5);
    int lane = (int)(threadIdx.x & 31u);
    if (wave >= nEdges) return;

    long long f = edges[2 * (long long)wave + 0];
    long long t = edges[2 * (long long)wave + 1];

    float4 a = *((const float4*)(P1 + (size_t)f * D_OUT) + lane);
    float4 b = *((const float4*)(P2 + (size_t)t * D_OUT) + lane);
    float s = a.x * b.x + a.y * b.y + a.z * b.z + a.w * b.w;

#pragma unroll
    for (int off = 16; off > 0; off >>= 1)
        s += __shfl_xor(s, off, 32);

    if (lane == 0) out[wave] = s;
}

extern "C" void kernel_launch(void* const* d_in, const int* in_sizes, int n_in,
                              void* d_out, int out_size, void* d_ws, size_t ws_size,
                              hipStream_t stream) {
    const float*     emb1  = (const float*)d_in[0];
    const float*     emb2  = (const float*)d_in[1];
    const long long* edges = (const long long*)d_in[2];   // int64 (N_EDGES, 2)
    const float*     W1    = (const float*)d_in[3];
    const float*     b1    = (const float*)d_in[4];
    const float*     W2    = (const float*)d_in[5];
    const float*     b2    = (const float*)d_in[6];
    float*           out   = (float*)d_out;

    const int M      = in_sizes[0] / D_IN;     // 100000 nodes
    const int nEdges = in_sizes[2] / 2;        // 1600000 edges

    // workspace layout
    float* P1  = (float*)d_ws;
    float* P2  = P1 + (size_t)M * D_OUT;
    float* WT1 = P2 + (size_t)M * D_OUT;
    float* WT2 = WT1 + (size_t)D_IN * D_OUT;

    // 1) transpose both weight matrices
    {
        int n = D_IN * D_OUT;
        int blk = 256, grd = (n + blk - 1) / blk;
        wt_transpose_kernel<<<grd, blk, 0, stream>>>(W1, WT1);
        wt_transpose_kernel<<<grd, blk, 0, stream>>>(W2, WT2);
    }

    // 2) WMMA projections (one wave per 16-row strip, 8 waves per block)
    {
        int strips = (M + 15) / 16;                       // 6250
        int blk = 256;
        int grd = (strips + 7) / 8;
        proj_gemm_wmma_kernel<<<grd, blk, 0, stream>>>(emb1, WT1, b1, P1, strips);
        proj_gemm_wmma_kernel<<<grd, blk, 0, stream>>>(emb2, WT2, b2, P2, strips);
    }

    // 3) per-edge gather + dot (one wave per edge)
    {
        int blk = 256;                                    // 8 edges per block
        long long totalThreads = (long long)nEdges * 32;
        int grd = (int)((totalThreads + blk - 1) / blk);
        edge_dot_kernel<<<grd, blk, 0, stream>>>(P1, P2, edges, out, nEdges);
    }
}